// MambaBlock_6923487282554
// MI455X (gfx1250) — compile-verified
//
#include <hip/hip_runtime.h>
#include <hip/hip_bf16.h>
#include <math.h>

typedef __attribute__((ext_vector_type(2))) float v2f;
typedef __attribute__((ext_vector_type(8))) float v8f;

#define D_MODEL 1024
#define D_INNER 2048
#define D_STATE 16
#define NPARAM  33   /* 2*D_STATE + 1 */

#define WMMA_F32(a, b, c) \
    c = __builtin_amdgcn_wmma_f32_16x16x4_f32(false, (a), false, (b), (short)0, (c), false, false)

// ---------------------------------------------------------------------------
// RMSNorm: one 256-thread block per row of (B*L, D)
// ---------------------------------------------------------------------------
__global__ __launch_bounds__(256)
void rmsnorm_kernel(const float* __restrict__ x, const float* __restrict__ w,
                    float* __restrict__ out, int D) {
    __shared__ float red[8];
    int row = blockIdx.x;
    const float* xr = x + (size_t)row * D;
    float ss = 0.f;
    for (int i = threadIdx.x; i < D; i += blockDim.x) { float v = xr[i]; ss += v * v; }
    #pragma unroll
    for (int off = 16; off > 0; off >>= 1) ss += __shfl_xor(ss, off);
    int wid = threadIdx.x >> 5;
    if ((threadIdx.x & 31) == 0) red[wid] = ss;
    __syncthreads();
    if (threadIdx.x < 8) {
        float t = red[threadIdx.x];
        #pragma unroll
        for (int off = 4; off > 0; off >>= 1) t += __shfl_xor(t, off);
        if (threadIdx.x == 0) red[0] = t;
    }
    __syncthreads();
    float scale = rsqrtf(red[0] / (float)D + 1e-6f);
    for (int i = threadIdx.x; i < D; i += blockDim.x)
        out[(size_t)row * D + i] = xr[i] * scale * w[i];
}

// ---------------------------------------------------------------------------
// C[M,N] = A[M,K] * W[N,K]^T (+ optional add[M,N]).
// One wave32 per 32x64 macro-tile = 2x4 grid of 16x16 fp32 WMMA tiles.
// Out-of-range rows/cols are handled by CLAMPING the load address (garbage
// only lands in tile elements that are never stored), so the inner loop is
// branch-free: 6x global_load_b64 + 8x v_wmma, no EXEC manipulation.
// Fragment layouts (ISA 7.12.2):
//   A frag (16x4): lanes 0-15 -> M=lane, K=k..k+1 ; lanes 16-31 -> K=k+2..k+3
//   B frag (4x16): lanes 0-15 -> N=lane, K=k..k+1 ; lanes 16-31 -> K=k+2..k+3
//   C/D (16x16):   vgpr v, lanes 0-15 -> row v, lanes 16-31 -> row v+8
// ---------------------------------------------------------------------------
__global__ __launch_bounds__(256)
void gemm_f32_wmma(const float* __restrict__ A, const float* __restrict__ W,
                   const float* __restrict__ add, float* __restrict__ C,
                   int M, int N, int K) {
    const int lane   = threadIdx.x & 31;
    const int waveId = blockIdx.x * (blockDim.x >> 5) + (threadIdx.x >> 5);
    const int tilesN = (N + 63) >> 6;            // 64-wide macro tiles
    const int tM     = (waveId / tilesN) << 5;   // 32-tall macro tiles
    const int tN     = (waveId % tilesN) << 6;
    if (tM >= M) return;

    const int half = lane >> 4;                  // which K pair this lane carries
    const int l15  = lane & 15;

    const int rA0 = min(tM + l15,      M - 1);   // clamped load rows
    const int rA1 = min(tM + 16 + l15, M - 1);
    const int rW0 = min(tN + l15,      N - 1);
    const int rW1 = min(tN + 16 + l15, N - 1);
    const int rW2 = min(tN + 32 + l15, N - 1);
    const int rW3 = min(tN + 48 + l15, N - 1);

    const float* a0p = A + (size_t)rA0 * K + 2 * half;
    const float* a1p = A + (size_t)rA1 * K + 2 * half;
    const float* b0p = W + (size_t)rW0 * K + 2 * half;
    const float* b1p = W + (size_t)rW1 * K + 2 * half;
    const float* b2p = W + (size_t)rW2 * K + 2 * half;
    const float* b3p = W + (size_t)rW3 * K + 2 * half;

    v8f acc00 = {}, acc01 = {}, acc02 = {}, acc03 = {};
    v8f acc10 = {}, acc11 = {}, acc12 = {}, acc13 = {};
    #pragma unroll 2
    for (int k = 0; k < K; k += 4) {
        v2f a0 = *(const v2f*)(a0p + k);
        v2f a1 = *(const v2f*)(a1p + k);
        v2f b0 = *(const v2f*)(b0p + k);
        v2f b1 = *(const v2f*)(b1p + k);
        v2f b2 = *(const v2f*)(b2p + k);
        v2f b3 = *(const v2f*)(b3p + k);
        WMMA_F32(a0, b0, acc00);
        WMMA_F32(a0, b1, acc01);
        WMMA_F32(a0, b2, acc02);
        WMMA_F32(a0, b3, acc03);
        WMMA_F32(a1, b0, acc10);
        WMMA_F32(a1, b1, acc11);
        WMMA_F32(a1, b2, acc12);
        WMMA_F32(a1, b3, acc13);
    }

    const int c0 = tN + l15;
    const int c1 = tN + 16 + l15;
    const int c2 = tN + 32 + l15;
    const int c3 = tN + 48 + l15;
    #pragma unroll
    for (int v = 0; v < 8; ++v) {
        int r0 = tM + v + half * 8;
        int r1 = r0 + 16;
        if (r0 < M) {
            if (c0 < N) { float t = acc00[v]; if (add) t += add[(size_t)r0 * N + c0]; C[(size_t)r0 * N + c0] = t; }
            if (c1 < N) { float t = acc01[v]; if (add) t += add[(size_t)r0 * N + c1]; C[(size_t)r0 * N + c1] = t; }
            if (c2 < N) { float t = acc02[v]; if (add) t += add[(size_t)r0 * N + c2]; C[(size_t)r0 * N + c2] = t; }
            if (c3 < N) { float t = acc03[v]; if (add) t += add[(size_t)r0 * N + c3]; C[(size_t)r0 * N + c3] = t; }
        }
        if (r1 < M) {
            if (c0 < N) { float t = acc10[v]; if (add) t += add[(size_t)r1 * N + c0]; C[(size_t)r1 * N + c0] = t; }
            if (c1 < N) { float t = acc11[v]; if (add) t += add[(size_t)r1 * N + c1]; C[(size_t)r1 * N + c1] = t; }
            if (c2 < N) { float t = acc12[v]; if (add) t += add[(size_t)r1 * N + c2]; C[(size_t)r1 * N + c2] = t; }
            if (c3 < N) { float t = acc13[v]; if (add) t += add[(size_t)r1 * N + c3]; C[(size_t)r1 * N + c3] = t; }
        }
    }
}

// ---------------------------------------------------------------------------
// Depthwise causal conv (k=4) + bias + SiLU.  Reads x_ssm half of xz
// (stride 2*D_INNER), writes (B,L,D_INNER) densely.
// ---------------------------------------------------------------------------
__global__ __launch_bounds__(256)
void conv_silu_kernel(const float* __restrict__ xz, const float* __restrict__ cw,
                      const float* __restrict__ cb, float* __restrict__ xc,
                      int B, int L, int D, int xzStride) {
    size_t idx = (size_t)blockIdx.x * blockDim.x + threadIdx.x;
    size_t total = (size_t)B * L * D;
    if (idx >= total) return;
    int d = (int)(idx % D);
    int l = (int)((idx / D) % L);
    int b = (int)(idx / ((size_t)D * L));
    float acc = cb[d];
    #pragma unroll
    for (int j = 0; j < 4; ++j) {
        int ls = l - 3 + j;
        if (ls >= 0)
            acc += cw[d * 4 + j] * xz[((size_t)b * L + ls) * xzStride + d];
    }
    float s = acc / (1.f + __expf(-acc));   // SiLU
    xc[idx] = s;
}

// ---------------------------------------------------------------------------
// Selective scan, fused softplus(delta) / A=-exp(A_log) / D-skip / silu(z)
// gate.  One channel per half-wave: lane&15 = state index n.  Sequential over
// L; y reduction via 4 shfl_xor steps (stays inside the 16-lane half).
// ---------------------------------------------------------------------------
__global__ __launch_bounds__(256)
void scan_kernel(const float* __restrict__ xc,    // (B,L,D) conv+silu
                 const float* __restrict__ ssm,   // (B,L,33)
                 const float* __restrict__ xz,    // (B,L,4096) for z
                 const float* __restrict__ A_log, // (D,16)
                 const float* __restrict__ Dsk,   // (D)
                 const float* __restrict__ W_dt,  // (D)
                 const float* __restrict__ b_dt,  // (D)
                 float* __restrict__ y,           // (B,L,D) gated output
                 int B, int L, int D, int zStride, int zOff) {
    const int lane = threadIdx.x & 31;
    const int wave = blockIdx.x * (blockDim.x >> 5) + (threadIdx.x >> 5);
    const int chan = wave * 2 + (lane >> 4);     // [0, B*D)
    const int n    = lane & 15;
    const int b    = chan / D;
    const int d    = chan % D;
    if (b >= B) return;

    const float Acoef = -__expf(A_log[(size_t)d * D_STATE + n]);
    const float wdt = W_dt[d], bdt = b_dt[d], dskip = Dsk[d];

    float h = 0.f;
    for (int l = 0; l < L; ++l) {
        size_t t = (size_t)b * L + l;
        float Bt  = ssm[t * NPARAM + n];
        float Ct  = ssm[t * NPARAM + D_STATE + n];
        float dtr = ssm[t * NPARAM + 2 * D_STATE];
        float xt  = xc[t * D + d];
        float dl  = dtr * wdt + bdt;
        float delta = (dl > 20.f) ? dl : __logf(1.f + __expf(dl));  // softplus
        float dA = __expf(delta * Acoef);
        h = dA * h + (delta * xt) * Bt;
        float p = h * Ct;
        p += __shfl_xor(p, 1);
        p += __shfl_xor(p, 2);
        p += __shfl_xor(p, 4);
        p += __shfl_xor(p, 8);
        if (n == 0) {
            float zt = xz[t * zStride + zOff + d];
            float g  = zt / (1.f + __expf(-zt));                    // silu(z)
            y[t * D + d] = (p + xt * dskip) * g;
        }
    }
}

// ---------------------------------------------------------------------------
extern "C" void kernel_launch(void* const* d_in, const int* in_sizes, int n_in,
                              void* d_out, int out_size, void* d_ws, size_t ws_size,
                              hipStream_t stream) {
    const float* x      = (const float*)d_in[0];
    const float* norm_w = (const float*)d_in[1];
    const float* W_in   = (const float*)d_in[2];
    const float* conv_w = (const float*)d_in[3];
    const float* conv_b = (const float*)d_in[4];
    const float* W_x    = (const float*)d_in[5];
    const float* A_log  = (const float*)d_in[6];
    const float* D_skip = (const float*)d_in[7];
    const float* W_dt   = (const float*)d_in[8];
    const float* b_dt   = (const float*)d_in[9];
    const float* W_out  = (const float*)d_in[10];
    float* out = (float*)d_out;

    const int Bb = 2, L = 1024;
    const int Rows = Bb * L;               // 2048 token rows

    // Workspace layout (floats): ~76 MB total
    float* ws  = (float*)d_ws;
    float* xn  = ws;                                   // 2048*1024
    float* xz  = xn  + (size_t)Rows * D_MODEL;         // 2048*4096
    float* xc  = xz  + (size_t)Rows * 2 * D_INNER;     // 2048*2048
    float* ssm = xc  + (size_t)Rows * D_INNER;         // 2048*33
    float* yv  = ssm + (size_t)Rows * NPARAM;          // 2048*2048

    // 1) RMSNorm
    rmsnorm_kernel<<<Rows, 256, 0, stream>>>(x, norm_w, xn, D_MODEL);

    // 2) xz = xn @ W_in^T   (2048x1024 x 1024x4096)
    {
        int M = Rows, N = 2 * D_INNER, K = D_MODEL;
        int tiles = ((M + 31) / 32) * ((N + 63) / 64);
        gemm_f32_wmma<<<(tiles + 7) / 8, 256, 0, stream>>>(xn, W_in, nullptr, xz, M, N, K);
    }

    // 3) depthwise conv + SiLU on the x_ssm half of xz
    {
        size_t total = (size_t)Rows * D_INNER;
        int blocks = (int)((total + 255) / 256);
        conv_silu_kernel<<<blocks, 256, 0, stream>>>(xz, conv_w, conv_b, xc,
                                                     Bb, L, D_INNER, 2 * D_INNER);
    }

    // 4) ssm_params = x_conv @ W_x^T   (N=33, tile-guarded via clamping)
    {
        int M = Rows, N = NPARAM, K = D_INNER;
        int tiles = ((M + 31) / 32) * ((N + 63) / 64);
        gemm_f32_wmma<<<(tiles + 7) / 8, 256, 0, stream>>>(xc, W_x, nullptr, ssm, M, N, K);
    }

    // 5) selective scan + gate (silu(z)) fused
    {
        int waves = Bb * D_INNER / 2;      // 2048 waves, 2 channels each
        scan_kernel<<<waves / 8, 256, 0, stream>>>(xc, ssm, xz, A_log, D_skip,
                                                   W_dt, b_dt, yv,
                                                   Bb, L, D_INNER,
                                                   2 * D_INNER, D_INNER);
    }

    // 6) out = y @ W_out^T + residual
    {
        int M = Rows, N = D_MODEL, K = D_INNER;
        int tiles = ((M + 31) / 32) * ((N + 63) / 64);
        gemm_f32_wmma<<<(tiles + 7) / 8, 256, 0, stream>>>(yv, W_out, x, out, M, N, K);
    }
}